// XRouter_71889162600594
// MI455X (gfx1250) — compile-verified
//
#include <hip/hip_runtime.h>
#include <hip/hip_bf16.h>
#include <math.h>

// Problem constants (match reference)
#define BS    4
#define SEQ   4096
#define IN_D  4096
#define NSK   8
#define EPSLN 1e-5f

typedef __attribute__((ext_vector_type(2))) float v2f;
typedef __attribute__((ext_vector_type(8))) float v8f;

// ---------------------------------------------------------------------------
// ws layout (floats): sums[BS][IN_D] | counts[BS][IN_D]   (total 32768 f32)
// ---------------------------------------------------------------------------

__global__ __launch_bounds__(256) void xrouter_zero_ws(float* __restrict__ ws) {
    ws[blockIdx.x * 256 + threadIdx.x] = 0.0f;
}

// Streaming masked reduction over seq. Grid: (IN_D/1024, SEQ/64, BS).
// Each thread owns 4 contiguous d (float4, coalesced), loops 64 seq rows,
// skips rows where mask==0 (uniform branch -> those rows are never read).
__global__ __launch_bounds__(256) void xrouter_reduce(
    const float* __restrict__ x,
    const int*   __restrict__ pad,
    const int*   __restrict__ inst,
    float*       __restrict__ ws)
{
    const int b  = blockIdx.z;
    const int d0 = blockIdx.x * 1024 + threadIdx.x * 4;
    const int s0 = blockIdx.y * 64;

    const int*   pm = pad  + b * SEQ;
    const int*   im = inst + b * SEQ;
    const float* xb = x + (size_t)b * SEQ * IN_D + d0;

    float4 sum = make_float4(0.f, 0.f, 0.f, 0.f);
    float4 cnt = make_float4(0.f, 0.f, 0.f, 0.f);

    for (int s = s0; s < s0 + 64; ++s) {
        if (pm[s] * im[s]) {                       // block-uniform branch
            float4 v = *(const float4*)(xb + (size_t)s * IN_D);
            sum.x += v.x; sum.y += v.y; sum.z += v.z; sum.w += v.w;
            cnt.x += (v.x != 0.f) ? 1.f : 0.f;     // ref counts x_rout != 0
            cnt.y += (v.y != 0.f) ? 1.f : 0.f;     // element-wise
            cnt.z += (v.z != 0.f) ? 1.f : 0.f;
            cnt.w += (v.w != 0.f) ? 1.f : 0.f;
        }
    }

    float* sums = ws + b * IN_D + d0;
    float* cnts = ws + BS * IN_D + b * IN_D + d0;
    atomicAdd(&sums[0], sum.x); atomicAdd(&sums[1], sum.y);
    atomicAdd(&sums[2], sum.z); atomicAdd(&sums[3], sum.w);
    atomicAdd(&cnts[0], cnt.x); atomicAdd(&cnts[1], cnt.y);
    atomicAdd(&cnts[2], cnt.z); atomicAdd(&cnts[3], cnt.w);
}

// ---------------------------------------------------------------------------
// Finalize: layernorms + 4x8x4096 GEMM via V_WMMA_F32_16X16X4_F32 + softmax.
// One block, 12 waves, ~213KB dynamic LDS (CDNA5 WGP has 320KB).
// ---------------------------------------------------------------------------
__global__ __launch_bounds__(384) void xrouter_finalize(
    const float* __restrict__ ws,
    const float* __restrict__ ln_in_w, const float* __restrict__ ln_in_b,
    const float* __restrict__ ln_w_w,  const float* __restrict__ ln_w_b,
    const float* __restrict__ ff_w,    const float* __restrict__ ff_b,
    float*       __restrict__ out)
{
    extern __shared__ float smem[];
    float* hn   = smem;                          // [BS][IN_D]  normalized h
    float* wn   = smem + BS * IN_D;              // [NSK][IN_D] normalized W
    float* zrow = smem + (BS + NSK) * IN_D;      // [IN_D] zeros (pad lanes)
    float* sl   = smem + (BS + NSK + 1) * IN_D;  // [BS][16] logit accum

    const int tid  = threadIdx.x;
    const int lane = tid & 31;
    const int wv   = tid >> 5;                   // wave id, 0..11

    if (tid < BS * 16) sl[tid] = 0.0f;
    for (int i = tid; i < IN_D; i += 384) zrow[i] = 0.0f;

    // ---- Phase A: per-wave row load + LN stats (wave32 shfl allreduce) ----
    {
        float* row;
        const float *gw, *gb;
        float s = 0.f, ss = 0.f;
        if (wv < BS) {
            row = hn + wv * IN_D; gw = ln_in_w; gb = ln_in_b;
            const float* sums = ws + wv * IN_D;
            const float* cnts = ws + BS * IN_D + wv * IN_D;
            for (int d = lane; d < IN_D; d += 32) {
                float v = sums[d] / cnts[d];        // masked mean over seq
                row[d] = v; s += v; ss += v * v;
            }
        } else {
            row = wn + (wv - BS) * IN_D; gw = ln_w_w; gb = ln_w_b;
            const float* fw = ff_w + (wv - BS) * IN_D;
            for (int d = lane; d < IN_D; d += 32) {
                float v = fw[d];
                row[d] = v; s += v; ss += v * v;
            }
        }
        for (int off = 16; off > 0; off >>= 1) {
            s  += __shfl_xor(s,  off, 32);
            ss += __shfl_xor(ss, off, 32);
        }
        const float mu   = s * (1.0f / IN_D);
        const float var  = ss * (1.0f / IN_D) - mu * mu;
        const float rstd = rsqrtf(var + EPSLN);
        for (int d = lane; d < IN_D; d += 32)
            row[d] = (row[d] - mu) * rstd * gw[d] + gb[d];
    }
    __syncthreads();

    // ---- Phase B: logits = h . W^T via f32 WMMA (M=batch, N=skill, K=d) ----
    // A 16x4 f32 layout (ISA 7.12.2): V0 = K0 (lanes 0-15) / K2 (lanes 16-31),
    //                                 V1 = K1 / K3.  B 4x16 assumed mirrored.
    // Invalid M rows / N cols read a dedicated LDS zero row: loads are
    // unconditional ds_load_b64 (8B aligned since k0 is even) -> no EXEC
    // divergence anywhere around the WMMA.
    if (wv < 8) {                              // wave-uniform -> EXEC stays full
        v8f acc = {0.f, 0.f, 0.f, 0.f, 0.f, 0.f, 0.f, 0.f};
        const int lm = lane & 15;
        const int hi = lane >> 4;              // 0 -> K{0,1}, 1 -> K{2,3}
        const float* arow = (lm < BS)  ? (hn + lm * IN_D) : zrow;
        const float* brow = (lm < NSK) ? (wn + lm * IN_D) : zrow;

        const int dbeg = wv * (IN_D / 8);
        for (int d = dbeg; d < dbeg + IN_D / 8; d += 4) {
            const int k0 = d + 2 * hi;
            v2f a  = *(const v2f*)(arow + k0);   // ds_load_b64
            v2f bb = *(const v2f*)(brow + k0);   // ds_load_b64
            acc = __builtin_amdgcn_wmma_f32_16x16x4_f32(
                false, a, false, bb, (short)0, acc, false, false);
        }
        // C layout: VGPR j, lanes 0-15 -> (M=j, N=lane). Batches live in j=0..3.
        if (lane < 16) {
            for (int j = 0; j < BS; ++j)
                atomicAdd(&sl[j * 16 + lane], acc[j]);
        }
    }
    __syncthreads();

    // ---- Phase C: bias + softmax over 8 skills, write [4,1,8] + aux ----
    if (tid < BS) {
        float l[NSK];
        float mx = -1e30f;
        for (int k = 0; k < NSK; ++k) {
            l[k] = sl[tid * 16 + k] + ff_b[k];
            mx = fmaxf(mx, l[k]);
        }
        float den = 0.f;
        for (int k = 0; k < NSK; ++k) { l[k] = expf(l[k] - mx); den += l[k]; }
        const float inv = 1.0f / den;
        for (int k = 0; k < NSK; ++k) out[tid * NSK + k] = l[k] * inv;
    }
    if (tid == BS) out[BS * NSK] = 0.0f;       // aux_loss
}

// ---------------------------------------------------------------------------
extern "C" void kernel_launch(void* const* d_in, const int* in_sizes, int n_in,
                              void* d_out, int out_size, void* d_ws, size_t ws_size,
                              hipStream_t stream) {
    const float* x        = (const float*)d_in[0];
    const int*   pad      = (const int*)  d_in[1];
    const int*   inst     = (const int*)  d_in[2];
    const float* ln_in_w  = (const float*)d_in[3];
    const float* ln_in_b  = (const float*)d_in[4];
    const float* ln_w_w   = (const float*)d_in[5];
    const float* ln_w_b   = (const float*)d_in[6];
    const float* ff_w     = (const float*)d_in[7];
    const float* ff_b     = (const float*)d_in[8];
    float*       out      = (float*)d_out;
    float*       ws       = (float*)d_ws;

    // 1) zero sums/counts workspace (2*BS*IN_D floats = 128KB)
    xrouter_zero_ws<<<(2 * BS * IN_D) / 256, 256, 0, stream>>>(ws);

    // 2) masked streaming reduction over seq (HBM-bound; skips mask==0 rows)
    dim3 grid(IN_D / 1024, SEQ / 64, BS);
    xrouter_reduce<<<grid, 256, 0, stream>>>(x, pad, inst, ws);

    // 3) layernorms + WMMA GEMM + softmax (single WGP, 213KB LDS)
    const size_t smem_bytes = ((BS + NSK + 1) * IN_D + BS * 16) * sizeof(float);
    xrouter_finalize<<<1, 384, smem_bytes, stream>>>(
        ws, ln_in_w, ln_in_b, ln_w_w, ln_w_b, ff_w, ff_b, out);
}